// MinimaxKnnHeadFast_8546984919095
// MI455X (gfx1250) — compile-verified
//
#include <hip/hip_runtime.h>
#include <hip/hip_bf16.h>
#include <stdint.h>

// ---------------------------------------------------------------------------
// MinimaxKnnHeadFast for MI455X (gfx1250)
//   K1: KL(dist) over [B,C] — TDM tensor loads (dense tile for stu, gather
//       mode for teacher rows) into padded LDS, f32 WMMA row-sum reduction.
//   K2: 4x atomicMax_u64 knockout passes (segmented top-4, exact tie order).
//   K3: decode winner, re-gather teacher row, emit outputs.
// ---------------------------------------------------------------------------

#define B_N 200000
#define S_N 25000
#define E_N 50000
#define A_N 8
#define C_N 128
#define ROW_DW 132                 // padded LDS row pitch (128 data + 4 pad dwords)
#define WAVES 4
#define ROWS_PER_WAVE 16
#define ROWS_PER_BLOCK (WAVES * ROWS_PER_WAVE)   // 64; B_N % 64 == 0

typedef unsigned int u32;
typedef unsigned long long u64;
typedef __attribute__((ext_vector_type(2))) float v2f;
typedef __attribute__((ext_vector_type(8))) float v8f;
typedef __attribute__((ext_vector_type(4))) u32  v4u;
typedef __attribute__((ext_vector_type(4))) int  v4i;
typedef __attribute__((ext_vector_type(8))) int  v8i;

#define HAS_TDM  __has_builtin(__builtin_amdgcn_tensor_load_to_lds)
#define HAS_WMMA __has_builtin(__builtin_amdgcn_wmma_f32_16x16x4_f32)

__device__ __forceinline__ float fast_rcp(float x) {
#if __has_builtin(__builtin_amdgcn_rcpf)
  return __builtin_amdgcn_rcpf(x);   // v_rcp_f32
#else
  return 1.0f / x;
#endif
}

__device__ __forceinline__ void kl_acc(float q, float p, float mq, float mp,
                                       float& zq, float& zp, float& t1) {
  const float aa = (p - mp) * 0.5f;   // 1/TEMP = 0.5
  const float bb = (q - mq) * 0.5f;
  const float ea = __expf(aa);
  zq += __expf(bb);
  zp += ea;
  t1 += ea * (aa - bb);
}

__global__ __launch_bounds__(128) void k_dist(
    const int* __restrict__ nn_mask, const int* __restrict__ ex_idx,
    const float* __restrict__ stu, const int* __restrict__ aug_idx,
    const float* __restrict__ teacher, float* __restrict__ dist)
{
  __shared__ __align__(16) float lds[WAVES * 2 * ROWS_PER_WAVE * ROW_DW];

  const int lane = (int)(threadIdx.x & 31u);
  const int wave = __builtin_amdgcn_readfirstlane((int)(threadIdx.x >> 5));
  const int t0   = (int)blockIdx.x * ROWS_PER_BLOCK + wave * ROWS_PER_WAVE;

  float* stuT = &lds[wave * (2 * ROWS_PER_WAVE * ROW_DW)];
  float* teaT = stuT + ROWS_PER_WAVE * ROW_DW;

  // w0 of D# group1: data_size=4B, pad_enable, pad_interval=128dw, pad_amount=4dw
  const int DESC_W0 = (2 << 16) | (1 << 20) | (6 << 22) | (3 << 25);

#if HAS_TDM
  // ---- TDM: dense 16x128 f32 tile of stu_logits -> LDS (padded rows) ----
  {
    const u32 ldsStu = (u32)(uintptr_t)stuT;
    const uint64_t ga = (uint64_t)(uintptr_t)(stu + (size_t)t0 * C_N);
    v4u g0;
    g0[0] = 1u;                                   // count=1, no gather
    g0[1] = ldsStu;                               // lds_addr
    g0[2] = (u32)ga;                              // global_addr[31:0]
    g0[3] = (u32)((ga >> 32) & 0x1FFFFFFu) | (2u << 30);  // addr[56:32] | type=2
    const u32 rows_rem = (u32)(B_N - t0);
    v8i g1;
    g1[0] = DESC_W0;
    g1[1] = (int)(((u32)C_N & 0xFFFFu) << 16);            // tensor_dim0 lo16
    g1[2] = (int)(((rows_rem & 0xFFFFu) << 16));          // td0 hi16=0 | td1 lo16
    g1[3] = (int)(((u32)C_N << 16) | ((rows_rem >> 16) & 0xFFFFu)); // td1 hi | tile_dim0
    g1[4] = ROWS_PER_WAVE;                                // tile_dim1=16, tile_dim2=0
    g1[5] = C_N;                                          // tensor_dim0_stride=128
    g1[6] = 0;
    g1[7] = 0;
    v4i z4 = {0, 0, 0, 0};
    v8i z8 = {0, 0, 0, 0, 0, 0, 0, 0};
    __builtin_amdgcn_tensor_load_to_lds(g0, g1, z4, z4, z8, 0);
  }
#endif

  // ---- per-row teacher gather indices (lanes 16..31 mirror lanes 0..15) ----
  const int m16 = lane & 15;
  const int b   = t0 + m16;
  const int seg = nn_mask[b];
  const int e   = ex_idx[seg];
  const int a   = aug_idx[b];
  const u32 rowIdx = (u32)e * (u32)A_N + (u32)a;   // row in the [E*A, C] matrix

#if HAS_TDM
  // ---- TDM gather mode: 2 descriptors x 8 teacher rows (32-bit indices) ----
  {
    const u32 ldsTea = (u32)(uintptr_t)teaT;
    const uint64_t ga = (uint64_t)(uintptr_t)teacher;
    v4u h0;
    h0[0] = 1u | (1u << 30) | (1u << 31);          // count=1, 32-bit idx, gather
    h0[1] = ldsTea;
    h0[2] = (u32)ga;
    h0[3] = (u32)((ga >> 32) & 0x1FFFFFFu) | (2u << 30);
    const u32 td1 = (u32)(E_N * A_N);              // 400000 rows
    v8i h1;
    h1[0] = DESC_W0;
    h1[1] = (int)(((u32)C_N & 0xFFFFu) << 16);
    h1[2] = (int)((td1 & 0xFFFFu) << 16);
    h1[3] = (int)(((u32)C_N << 16) | ((td1 >> 16) & 0xFFFFu));
    h1[4] = 8;                                     // tile_dim1 = #valid indices
    h1[5] = C_N;                                   // tensor_dim0_stride
    h1[6] = 0;
    h1[7] = 0;
    v8i z8 = {0, 0, 0, 0, 0, 0, 0, 0};
    v4i i0, i1, j0, j1;
    i0[0] = __builtin_amdgcn_readlane((int)rowIdx, 0);
    i0[1] = __builtin_amdgcn_readlane((int)rowIdx, 1);
    i0[2] = __builtin_amdgcn_readlane((int)rowIdx, 2);
    i0[3] = __builtin_amdgcn_readlane((int)rowIdx, 3);
    i1[0] = __builtin_amdgcn_readlane((int)rowIdx, 4);
    i1[1] = __builtin_amdgcn_readlane((int)rowIdx, 5);
    i1[2] = __builtin_amdgcn_readlane((int)rowIdx, 6);
    i1[3] = __builtin_amdgcn_readlane((int)rowIdx, 7);
    __builtin_amdgcn_tensor_load_to_lds(h0, h1, i0, i1, z8, 0);

    j0[0] = __builtin_amdgcn_readlane((int)rowIdx, 8);
    j0[1] = __builtin_amdgcn_readlane((int)rowIdx, 9);
    j0[2] = __builtin_amdgcn_readlane((int)rowIdx, 10);
    j0[3] = __builtin_amdgcn_readlane((int)rowIdx, 11);
    j1[0] = __builtin_amdgcn_readlane((int)rowIdx, 12);
    j1[1] = __builtin_amdgcn_readlane((int)rowIdx, 13);
    j1[2] = __builtin_amdgcn_readlane((int)rowIdx, 14);
    j1[3] = __builtin_amdgcn_readlane((int)rowIdx, 15);
    h0[1] = ldsTea + (u32)(8 * ROW_DW * 4);
    __builtin_amdgcn_tensor_load_to_lds(h0, h1, j0, j1, z8, 0);
  }
  #if __has_builtin(__builtin_amdgcn_s_wait_tensorcnt)
  __builtin_amdgcn_s_wait_tensorcnt(0);
  #else
  asm volatile("s_wait_tensorcnt 0" ::: "memory");
  #endif
#else
  // ---- fallback staging: plain global loads + LDS stores (same layout) ----
  #pragma unroll
  for (int r = 0; r < ROWS_PER_WAVE; ++r) {
    float4 sq = ((const float4*)(stu + (size_t)(t0 + r) * C_N))[lane];
    ((float4*)(stuT + r * ROW_DW))[lane] = sq;
    u32 ro = (u32)__builtin_amdgcn_readlane((int)rowIdx, r);
    float4 tq = ((const float4*)((const char*)teacher + (size_t)ro * (C_N * 4)))[lane];
    ((float4*)(teaT + r * ROW_DW))[lane] = tq;
  }
#endif

  // ---- compute: lane = (row m16, half h); 64 columns per lane ----
  const int h = lane >> 4;
  const float* qrow = stuT + m16 * ROW_DW + h * 64;
  const float* prow = teaT + m16 * ROW_DW + h * 64;

  float mq = -3.0e38f, mp = -3.0e38f;
  #pragma unroll
  for (int j = 0; j < 16; ++j) {
    float4 q = ((const float4*)qrow)[j];
    float4 p = ((const float4*)prow)[j];
    mq = fmaxf(mq, fmaxf(fmaxf(q.x, q.y), fmaxf(q.z, q.w)));
    mp = fmaxf(mp, fmaxf(fmaxf(p.x, p.y), fmaxf(p.z, p.w)));
  }
  mq = fmaxf(mq, __shfl_xor(mq, 16, 32));
  mp = fmaxf(mp, __shfl_xor(mp, 16, 32));

  float zq = 0.f, zp = 0.f, t1 = 0.f;
  #pragma unroll
  for (int j = 0; j < 16; ++j) {
    float4 q = ((const float4*)qrow)[j];
    float4 p = ((const float4*)prow)[j];
    kl_acc(q.x, p.x, mq, mp, zq, zp, t1);
    kl_acc(q.y, p.y, mq, mp, zq, zp, t1);
    kl_acc(q.z, p.z, mq, mp, zq, zp, t1);
    kl_acc(q.w, p.w, mq, mp, zq, zp, t1);
  }

#if HAS_WMMA
  // ---- row reduction on the matrix pipe: D = A(16x4 partials) x ones(4x16)
  // A layout: lane(m,h) supplies K=(2h) in v0, K=(2h+1) in v1; put partial in
  // v0, 0 in v1 -> row sum = partial(m,0)+partial(m,1). EXEC is all-1s here.
  v2f ones; ones.x = 1.0f; ones.y = 1.0f;
  v8f c0 = {0.f, 0.f, 0.f, 0.f, 0.f, 0.f, 0.f, 0.f};
  v2f Aq; Aq.x = zq; Aq.y = 0.f;
  v8f Dq = __builtin_amdgcn_wmma_f32_16x16x4_f32(false, Aq, false, ones,
                                                 (short)0, c0, false, false);
  v2f Ap; Ap.x = zp; Ap.y = 0.f;
  v8f Dp = __builtin_amdgcn_wmma_f32_16x16x4_f32(false, Ap, false, ones,
                                                 (short)0, c0, false, false);
  v2f At; At.x = t1; At.y = 0.f;
  v8f Dt = __builtin_amdgcn_wmma_f32_16x16x4_f32(false, At, false, ones,
                                                 (short)0, c0, false, false);
  // D layout: vgpr r, lanes 0-15 -> M=r (N=lane), lanes 16-31 -> M=r+8.
  // Lane 0 holds rows 0..7, lane 16 holds rows 8..15 (any N works; use N=0).
  if ((lane & 15) == 0) {
    const int rbase = t0 + h * 8;
    #pragma unroll
    for (int r = 0; r < 8; ++r) {
      float Zq = Dq[r], Zp = Dp[r], T1 = Dt[r];
      dist[rbase + r] = T1 * fast_rcp(Zp) - __logf(Zp) + __logf(Zq);
    }
  }
#else
  zq += __shfl_xor(zq, 16, 32);
  zp += __shfl_xor(zp, 16, 32);
  t1 += __shfl_xor(t1, 16, 32);
  if (lane < 16)
    dist[t0 + lane] = t1 * fast_rcp(zp) - __logf(zp) + __logf(zq);
#endif
}

// ---------------------------------------------------------------------------
__global__ void k_zero(u64* __restrict__ p, int n) {
  int i = (int)(blockIdx.x * blockDim.x + threadIdx.x);
  if (i < n) p[i] = 0ull;
}

// One knockout round: among keys strictly below previous round's winner,
// take the segment max. key = (orderedF32(dist)<<32) | (~idx) -> unique keys,
// ordering == (dist desc, idx asc), exactly the reference tie rule.
__global__ void k_pass(const float* __restrict__ dist, const int* __restrict__ seg,
                       const u64* __restrict__ prev, u64* __restrict__ cur, int first)
{
  int i = (int)(blockIdx.x * blockDim.x + threadIdx.x);
  if (i >= B_N) return;
  float d = dist[i];
  u32 u = __float_as_uint(d);
  u = (u & 0x80000000u) ? ~u : (u | 0x80000000u);
  u64 key = ((u64)u << 32) | (u64)(0xFFFFFFFFu - (u32)i);
  int s = seg[i];
  if (first || key < prev[s]) atomicMax(cur + s, key);
}

__global__ __launch_bounds__(128) void k_out(const u64* __restrict__ best4,
    const int* __restrict__ nn_mask, const int* __restrict__ ex_idx,
    const int* __restrict__ aug, const int* __restrict__ ranks,
    const float* __restrict__ teacher, float* __restrict__ out)
{
  const int s = (int)blockIdx.x;
  const u64 key = best4[s];                       // 0 iff segment had <4 elems
  const u32 hi = (u32)(key >> 32);
  const u32 vb = (hi & 0x80000000u) ? (hi ^ 0x80000000u) : ~hi;
  const float val = __uint_as_float(vb);
  const bool valid = (key != 0ull) && (val != 0.0f) && __builtin_isfinite(val);
  const int idx = (int)(0xFFFFFFFFu - (u32)(key & 0xFFFFFFFFull));
  const int sel = valid ? idx : 0;
  const int e = ex_idx[nn_mask[sel]];
  const int a = aug[sel];
  const float* trow = teacher + ((size_t)e * A_N + (size_t)a) * C_N;
  out[(size_t)s * C_N + threadIdx.x] = trow[threadIdx.x];
  if (threadIdx.x == 0) {
    float* o2 = out + (size_t)S_N * C_N;
    o2[s] = (float)sel;
    o2[S_N + s] = valid ? 1.0f : 0.0f;
    o2[2 * S_N + s] = (float)ranks[sel];
  }
}

// ---------------------------------------------------------------------------
extern "C" void kernel_launch(void* const* d_in, const int* in_sizes, int n_in,
                              void* d_out, int out_size, void* d_ws, size_t ws_size,
                              hipStream_t stream)
{
  (void)in_sizes; (void)n_in; (void)out_size; (void)ws_size;
  // inputs: 0 augment_rank(=4), 1 nn_mask[B], 2 example_indices[S],
  //         3 stu_logits[B*C], 4 augmented_indices[B], 5 nn_ranks[B],
  //         6 teacher_table[E*A*C]
  const int*   nn_mask = (const int*)d_in[1];
  const int*   ex_idx  = (const int*)d_in[2];
  const float* stu     = (const float*)d_in[3];
  const int*   aug     = (const int*)d_in[4];
  const int*   ranks   = (const int*)d_in[5];
  const float* teach   = (const float*)d_in[6];
  float* out = (float*)d_out;

  float* dist = (float*)d_ws;                                     // B floats
  u64* best = (u64*)((char*)d_ws + (size_t)B_N * sizeof(float));  // 4*S u64

  k_zero<<<(4 * S_N + 255) / 256, 256, 0, stream>>>(best, 4 * S_N);
  k_dist<<<B_N / ROWS_PER_BLOCK, WAVES * 32, 0, stream>>>(nn_mask, ex_idx, stu,
                                                          aug, teach, dist);
  for (int r = 0; r < 4; ++r) {
    k_pass<<<(B_N + 255) / 256, 256, 0, stream>>>(
        dist, nn_mask, best + (size_t)(r ? (r - 1) : 0) * S_N,
        best + (size_t)r * S_N, r == 0 ? 1 : 0);
  }
  k_out<<<S_N, C_N, 0, stream>>>(best + 3ull * S_N, nn_mask, ex_idx, aug, ranks,
                                 teach, out);
}